// Caps_Layer_30958124269624
// MI455X (gfx1250) — compile-verified
//
#include <hip/hip_runtime.h>

typedef __attribute__((ext_vector_type(2))) float v2f;
typedef __attribute__((ext_vector_type(8))) float v8f;

#define BATCH 128
#define SEQ   512
#define DIM   768
#define NCAP  5
#define KCAP  5
#define NK    25      // NCAP*KCAP (GEMM N)
#define NPAD  32      // N padded to two 16-wide WMMA tiles
#define KCHUNK 384    // K rows of W staged in LDS per pass (2 passes over K=768)
#define KSTRIDE 388   // padded row stride (388 % 64 == 4 -> conflict-free b64 reads)

// ---------------------------------------------------------------------------
// Phase 1: u_hat[M=B*S][25] = x[M][768] @ W[768][25], fp32 WMMA 16x16x4.
// Memory-bound on streaming x (201 MB @ 23.3 TB/s ~ 8.6 us floor).
// W staged TRANSPOSED in LDS: Wl[n][k], so each lane's B frag (K=k,k+1 at
// fixed n) is one aligned ds_load_b64 -> no VGPR repacking before WMMA.
// ---------------------------------------------------------------------------
__global__ __launch_bounds__(256)
void capsule_gemm_wmma(const float* __restrict__ x,
                       const float* __restrict__ W,
                       float* __restrict__ u_hat) {
  __shared__ float Wl[NPAD * KSTRIDE];   // ~48.5 KB

  const int tid  = threadIdx.x;
  const int lane = tid & 31;
  const int wave = tid >> 5;
  const int lrow = lane & 15;
  const int half = lane >> 4;           // 0: lanes 0-15, 1: lanes 16-31

  const int m_base = (blockIdx.x * 8 + wave) * 16;

  v8f acc0 = {};   // N tile 0: cols 0..15
  v8f acc1 = {};   // N tile 1: cols 16..31 (25..31 zero-padded, discarded)

  for (int ch = 0; ch < 2; ++ch) {
    const int kbase = ch * KCHUNK;
    // Cooperative zero-padded transpose-stage of W[kbase..kbase+KCHUNK) into
    // LDS. 12288 elements / 256 threads: uniform trip count, no divergence.
    for (int idx = tid; idx < KCHUNK * NPAD; idx += 256) {
      const int k = idx >> 5;
      const int n = idx & 31;
      Wl[n * KSTRIDE + k] = (n < NK) ? W[(size_t)(kbase + k) * NK + n] : 0.0f;
    }
    __syncthreads();

    // A-matrix 16x4 f32 layout: lanes 0-15 hold (K=k0,k0+1), lanes 16-31 hold
    // (K=k0+2,k0+3) for row M=lane%16 -> one aligned float2 per lane per step.
    const float2* arow =
        (const float2*)(x + (size_t)(m_base + lrow) * DIM + kbase + 2 * half);

    // B-matrix 4x16 f32 layout (mirror of A): lanes 0-15 -> K=k0,k0+1;
    // lanes 16-31 -> K=k0+2,k0+3; N = lane%16 (+16 for tile 1).
    // Transposed LDS rows make each frag one contiguous 8B-aligned pair.
    const v2f* brow0 = (const v2f*)&Wl[lrow * KSTRIDE + 2 * half];
    const v2f* brow1 = (const v2f*)&Wl[(16 + lrow) * KSTRIDE + 2 * half];

    #pragma unroll 8
    for (int kc = 0; kc < KCHUNK; kc += 4) {
      const float2 av = arow[kc >> 1];
      v2f a; a.x = av.x; a.y = av.y;
      const v2f b0 = brow0[kc >> 1];
      const v2f b1 = brow1[kc >> 1];

      acc0 = __builtin_amdgcn_wmma_f32_16x16x4_f32(
          false, a, false, b0, (short)0, acc0, false, false);
      acc1 = __builtin_amdgcn_wmma_f32_16x16x4_f32(
          false, a, false, b1, (short)0, acc1, false, false);
    }
    __syncthreads();
  }

  // D layout: VGPR r -> M = m_base + r (lanes 0-15) / m_base + r + 8 (16-31),
  // N = lane%16 (+16 for tile 1). Drop padded columns 25..31.
  #pragma unroll
  for (int r = 0; r < 8; ++r) {
    const int m = m_base + r + 8 * half;
    u_hat[(size_t)m * NK + lrow] = acc0[r];
    if (lrow + 16 < NK)
      u_hat[(size_t)m * NK + 16 + lrow] = acc1[r];
  }
}

// ---------------------------------------------------------------------------
// Phase 2: dynamic routing, one block per batch element. Each thread owns
// s = tid and s = tid+256: u_hat rows and logits live in registers; only the
// 25-value o[n][k] reduction goes through LDS (ds_add_f32 atomics).
// ---------------------------------------------------------------------------
__global__ __launch_bounds__(256)
void capsule_routing(const float* __restrict__ u_hat, float* __restrict__ out) {
  const int b   = blockIdx.x;
  const int tid = threadIdx.x;

  float u[2][NK];
  #pragma unroll
  for (int j = 0; j < 2; ++j) {
    const int s = tid + j * 256;
    const float* row = u_hat + (size_t)(b * SEQ + s) * NK;
    #pragma unroll
    for (int e = 0; e < NK; ++e) u[j][e] = row[e];
  }

  float bb[2][NCAP];
  #pragma unroll
  for (int j = 0; j < 2; ++j)
    #pragma unroll
    for (int n = 0; n < NCAP; ++n) bb[j][n] = 0.0f;

  __shared__ float o_s[NK];

  for (int it = 0; it < 4; ++it) {
    if (tid < NK) o_s[tid] = 0.0f;
    __syncthreads();

    float po[NK];
    #pragma unroll
    for (int e = 0; e < NK; ++e) po[e] = 0.0f;

    #pragma unroll
    for (int j = 0; j < 2; ++j) {
      // softmax over the n axis (thread-local, 5 entries)
      float m = bb[j][0];
      #pragma unroll
      for (int n = 1; n < NCAP; ++n) m = fmaxf(m, bb[j][n]);
      float e5[NCAP], sum = 0.0f;
      #pragma unroll
      for (int n = 0; n < NCAP; ++n) { e5[n] = __expf(bb[j][n] - m); sum += e5[n]; }
      const float inv = 1.0f / sum;
      #pragma unroll
      for (int n = 0; n < NCAP; ++n) {
        const float c = e5[n] * inv;
        #pragma unroll
        for (int k = 0; k < KCAP; ++k)
          po[n * KCAP + k] += c * u[j][n * KCAP + k];
      }
    }

    #pragma unroll
    for (int e = 0; e < NK; ++e) atomicAdd(&o_s[e], po[e]);
    __syncthreads();

    // squash: v / sqrt(sum(v^2) + 1e-7) over k, per capsule n
    if (tid < NCAP) {
      float ss = 1e-7f;
      #pragma unroll
      for (int k = 0; k < KCAP; ++k) {
        const float v = o_s[tid * KCAP + k];
        ss += v * v;
      }
      const float r = rsqrtf(ss);
      #pragma unroll
      for (int k = 0; k < KCAP; ++k) o_s[tid * KCAP + k] *= r;
    }
    __syncthreads();

    if (it < 3) {
      #pragma unroll
      for (int j = 0; j < 2; ++j)
        #pragma unroll
        for (int n = 0; n < NCAP; ++n) {
          float acc = 0.0f;
          #pragma unroll
          for (int k = 0; k < KCAP; ++k)
            acc += o_s[n * KCAP + k] * u[j][n * KCAP + k];
          bb[j][n] = acc;
        }
    }
    __syncthreads();  // protect o_s reads from next iteration's zeroing
  }

  if (tid < NK) out[b * NK + tid] = o_s[tid];
}

extern "C" void kernel_launch(void* const* d_in, const int* in_sizes, int n_in,
                              void* d_out, int out_size, void* d_ws, size_t ws_size,
                              hipStream_t stream) {
  const float* x = (const float*)d_in[0];   // [128, 512, 768] f32
  const float* W = (const float*)d_in[1];   // [768, 25] f32
  float* u_hat   = (float*)d_ws;            // [65536, 25] f32 = 6.55 MB scratch
  float* out     = (float*)d_out;           // [128, 5, 5] f32

  // 512 blocks * 8 waves * 16 rows = 65536 M rows
  capsule_gemm_wmma<<<512, 256, 0, stream>>>(x, W, u_hat);
  capsule_routing<<<BATCH, 256, 0, stream>>>(u_hat, out);
}